// TGNN_61383672594891
// MI455X (gfx1250) — compile-verified
//
#include <hip/hip_runtime.h>
#include <math.h>

typedef __attribute__((ext_vector_type(2))) float v2f;
typedef __attribute__((ext_vector_type(8))) float v8f;

#define THRES1 0.8f
#define THRES2 0.4f
#define RCOEF  1.0f

#define SCAN_BLOCK 256
#define SCAN_L     16
#define SCAN_CHUNK (SCAN_BLOCK * SCAN_L)   // 4096 elements per chunk

// ---------------------------------------------------------------------------
// Phase 1: a = sigmoid(h @ W + b) via V_WMMA_F32_16X16X4_F32.
// Each wave owns 16 rows of h. A-matrix tile = 16x4 f32 slab of h,
// B-matrix = the matching 4 W values broadcast across all 16 columns,
// so every column of D equals the row dot product; accumulate over K=feat.
// ---------------------------------------------------------------------------
__global__ void gemv_sigmoid_wmma(const float* __restrict__ h,
                                  const float* __restrict__ W,
                                  const float* __restrict__ bias_p,
                                  float* __restrict__ a_out,
                                  int seq, int feat, int ntiles) {
  __shared__ float wsh[512];
  for (int k = threadIdx.x; k < feat; k += blockDim.x) wsh[k] = W[k];
  __syncthreads();
  const float bias = bias_p[0];
  const int lane = threadIdx.x & 31;
  const int wave = threadIdx.x >> 5;
  const int wpb  = blockDim.x >> 5;
  const int m    = lane & 15;     // row within 16-row tile
  const int half = lane >> 4;     // selects K pair (ISA A-layout: VGPR0=K0/K2)
  const int stride = gridDim.x * wpb;
  for (int t = blockIdx.x * wpb + wave; t < ntiles; t += stride) {
    int row = t * 16 + m;
    if (row >= seq) row = seq - 1;                 // clamp (redundant compute)
    const float* hrow = h + (size_t)row * feat + 2 * half;
    v8f acc = {};
    for (int k0 = 0; k0 < feat; k0 += 4) {
      v2f av; av.x = hrow[k0];          av.y = hrow[k0 + 1];
      v2f bv; bv.x = wsh[k0 + 2*half];  bv.y = wsh[k0 + 2*half + 1];
      acc = __builtin_amdgcn_wmma_f32_16x16x4_f32(
          /*neg_a=*/false, av, /*neg_b=*/false, bv,
          /*c_mod=*/(short)0, acc, /*reuse_a=*/false, /*reuse_b=*/false);
    }
    if (m == 0) {                 // lane 0 -> rows 0..7, lane 16 -> rows 8..15
      int base = t * 16 + 8 * half;
#pragma unroll
      for (int j = 0; j < 8; ++j) {
        int r = base + j;
        if (r < seq) {
          float x = acc[j] + bias;
          a_out[r] = 1.0f / (1.0f + __expf(-x));
        }
      }
    }
  }
}

// ---------------------------------------------------------------------------
// Affine-recurrence parallel scan: x_i = m_i * x_{i-1} + c_i
// MODE 0: alpha scan   (m = up_i,         c = (1-up_i)*a_i)
// MODE 1: y scan       (m = g*(1-alpha),  c = g*alpha*up,  g = zeroing gate)
// ---------------------------------------------------------------------------
template<int MODE>
__device__ __forceinline__ void elem_coeff(long i, const float* __restrict__ p0,
                                           const float* __restrict__ up,
                                           const float* __restrict__ u,
                                           float& m, float& c) {
  float upi = up[i];
  if (MODE == 0) {
    m = upi;
    c = (1.0f - upi) * p0[i];
  } else {
    float al = p0[i];
    bool z = (upi <= 0.5f) || (u[i] > 1.0f);
    m = z ? 0.0f : (1.0f - al);
    c = z ? 0.0f : (al * upi);
  }
}

template<int MODE>
__global__ void scan_reduce(const float* __restrict__ p0,
                            const float* __restrict__ up,
                            const float* __restrict__ u,
                            float* __restrict__ chunkM,
                            float* __restrict__ chunkC, int seq) {
  __shared__ float sm[SCAN_BLOCK], sc[SCAN_BLOCK];
  const int t = threadIdx.x;
  const long base = (long)blockIdx.x * SCAN_CHUNK + (long)t * SCAN_L;
  float M = 1.0f, C = 0.0f;
  for (int j = 0; j < SCAN_L; ++j) {
    long i = base + j;
    float m = 1.0f, c = 0.0f;
    if (i < seq) elem_coeff<MODE>(i, p0, up, u, m, c);
    C = m * C + c;   // compose new element after current composition
    M = m * M;
  }
  sm[t] = M; sc[t] = C;
  __syncthreads();
  for (int off = 1; off < SCAN_BLOCK; off <<= 1) {
    float pm = 1.0f, pc = 0.0f;
    bool has = (t >= off);
    if (has) { pm = sm[t - off]; pc = sc[t - off]; }
    __syncthreads();
    if (has) { sc[t] = sm[t] * pc + sc[t]; sm[t] = sm[t] * pm; }
    __syncthreads();
  }
  if (t == SCAN_BLOCK - 1) { chunkM[blockIdx.x] = sm[t]; chunkC[blockIdx.x] = sc[t]; }
}

__global__ void scan_spine(const float* __restrict__ chunkM,
                           const float* __restrict__ chunkC,
                           float* __restrict__ chunkIn, int nchunks) {
  if (blockIdx.x == 0 && threadIdx.x == 0) {
    float x = 0.0f;                       // initial carry (alpha0 = y0 = 0)
    for (int j = 0; j < nchunks; ++j) {
      chunkIn[j] = x;
      x = chunkM[j] * x + chunkC[j];
    }
  }
}

template<int MODE>
__global__ void scan_apply(const float* __restrict__ p0,
                           const float* __restrict__ up,
                           const float* __restrict__ u,
                           const float* __restrict__ chunkIn,
                           float* __restrict__ out, int seq) {
  __shared__ float sm[SCAN_BLOCK], sc[SCAN_BLOCK];
  const int t = threadIdx.x;
  const long base = (long)blockIdx.x * SCAN_CHUNK + (long)t * SCAN_L;
  float M = 1.0f, C = 0.0f;
  for (int j = 0; j < SCAN_L; ++j) {
    long i = base + j;
    float m = 1.0f, c = 0.0f;
    if (i < seq) elem_coeff<MODE>(i, p0, up, u, m, c);
    C = m * C + c;
    M = m * M;
  }
  sm[t] = M; sc[t] = C;
  __syncthreads();
  for (int off = 1; off < SCAN_BLOCK; off <<= 1) {
    float pm = 1.0f, pc = 0.0f;
    bool has = (t >= off);
    if (has) { pm = sm[t - off]; pc = sc[t - off]; }
    __syncthreads();
    if (has) { sc[t] = sm[t] * pc + sc[t]; sm[t] = sm[t] * pm; }
    __syncthreads();
  }
  float x = chunkIn[blockIdx.x];
  if (t > 0) x = sm[t - 1] * x + sc[t - 1];   // exclusive prefix into this run
  for (int j = 0; j < SCAN_L; ++j) {
    long i = base + j;
    if (i < seq) {
      float m, c;
      elem_coeff<MODE>(i, p0, up, u, m, c);
      x = m * x + c;
      out[i] = x;
    }
  }
}

// ---------------------------------------------------------------------------
// Loss: trig positions depend only on `up` (falling edge or last index), and
// l_c/l_e/calc_flag reset at every trig -> segments are independent.
// One thread per element; trig threads walk their own (short) segment.
// Deterministic reduction: block tree + single-thread final sum.
// ---------------------------------------------------------------------------
__global__ void loss_kernel(const float* __restrict__ y,
                            const float* __restrict__ up,
                            const float* __restrict__ label,
                            float* __restrict__ out_up,
                            float* __restrict__ blockLoss,
                            float* __restrict__ blockCnt, int seq) {
  __shared__ float sl[256], sn[256];
  const int t = threadIdx.x;
  const int i = blockIdx.x * 256 + t;
  float add = 0.0f, cadd = 0.0f;
  if (i < seq) {
    float upi = up[i];
    out_up[i] = upi;                              // pass-through output
    float up_pre = (i > 0) ? up[i - 1] : 0.0f;
    bool trig = ((up_pre >= 0.5f) && (upi < 0.5f)) || (i == seq - 1);
    if (trig) {
      int start = 0;                               // previous trig + 1
      for (int j = i - 1; j > 0; --j) {
        if ((up[j - 1] >= 0.5f) && (up[j] < 0.5f)) { start = j + 1; break; }
      }
      float l_c = 0.0f, l_e = 0.0f;
      bool cf = false;
      for (int j = start; j <= i; ++j) {
        float yj = y[j];
        float lj = label[j];
        if (lj >= THRES1) { float d = yj - lj; l_c = d * d; }      // latch last
        if ((yj >= THRES1) && !cf) { float d = yj - THRES2; l_e = d * d; cf = true; }
      }
      add  = (l_c != 0.0f) ? l_c * RCOEF : ((l_e != 0.0f) ? l_e : 0.0f);
      cadd = ((l_c != 0.0f) || (l_e != 0.0f)) ? 1.0f : 0.0f;
    }
  }
  sl[t] = add; sn[t] = cadd;
  __syncthreads();
  for (int off = 128; off > 0; off >>= 1) {
    if (t < off) { sl[t] += sl[t + off]; sn[t] += sn[t + off]; }
    __syncthreads();
  }
  if (t == 0) { blockLoss[blockIdx.x] = sl[0]; blockCnt[blockIdx.x] = sn[0]; }
}

__global__ void finalize_loss(const float* __restrict__ bl,
                              const float* __restrict__ bc, int nb,
                              float* __restrict__ out_loss,
                              float* __restrict__ out_cnt) {
  if (blockIdx.x == 0 && threadIdx.x == 0) {
    float L = 0.0f, C = 0.0f;
    for (int i = 0; i < nb; ++i) { L += bl[i]; C += bc[i]; }
    *out_loss = L;
    *out_cnt = C;
  }
}

// ---------------------------------------------------------------------------
extern "C" void kernel_launch(void* const* d_in, const int* in_sizes, int n_in,
                              void* d_out, int out_size, void* d_ws, size_t ws_size,
                              hipStream_t stream) {
  const float* h     = (const float*)d_in[0];
  const float* u     = (const float*)d_in[1];
  const float* up    = (const float*)d_in[2];
  const float* label = (const float*)d_in[3];
  const float* W     = (const float*)d_in[4];
  const float* b     = (const float*)d_in[5];
  const int seq  = in_sizes[1];
  const int feat = in_sizes[4];

  float* out       = (float*)d_out;          // [y | alpha | up | loss | cnt]
  float* out_y     = out;
  float* out_alpha = out + (size_t)seq;
  float* out_up    = out + 2 * (size_t)seq;
  float* out_loss  = out + 3 * (size_t)seq;
  float* out_cnt   = out_loss + 1;

  const int nchunks = (seq + SCAN_CHUNK - 1) / SCAN_CHUNK;
  const int nlb     = (seq + 255) / 256;

  float* a_ws = (float*)d_ws;                // sigmoid activations [seq]
  float* chM  = a_ws + (size_t)seq;
  float* chC  = chM + nchunks;
  float* chIn = chC + nchunks;
  float* blL  = chIn + nchunks;
  float* blC  = blL + nlb;

  // 1) GEMV + sigmoid (WMMA f32)
  const int ntiles = (seq + 15) / 16;
  int gblocks = (ntiles + 7) / 8;            // 8 waves / 256-thread block
  if (gblocks > 4096) gblocks = 4096;
  gemv_sigmoid_wmma<<<gblocks, 256, 0, stream>>>(h, W, b, a_ws, seq, feat, ntiles);

  // 2) alpha scan
  scan_reduce<0><<<nchunks, SCAN_BLOCK, 0, stream>>>(a_ws, up, u, chM, chC, seq);
  scan_spine<<<1, 32, 0, stream>>>(chM, chC, chIn, nchunks);
  scan_apply<0><<<nchunks, SCAN_BLOCK, 0, stream>>>(a_ws, up, u, chIn, out_alpha, seq);

  // 3) y scan (coefficients depend on alpha)
  scan_reduce<1><<<nchunks, SCAN_BLOCK, 0, stream>>>(out_alpha, up, u, chM, chC, seq);
  scan_spine<<<1, 32, 0, stream>>>(chM, chC, chIn, nchunks);
  scan_apply<1><<<nchunks, SCAN_BLOCK, 0, stream>>>(out_alpha, up, u, chIn, out_y, seq);

  // 4) segmented loss + up pass-through, deterministic reduction
  loss_kernel<<<nlb, 256, 0, stream>>>(out_y, up, label, out_up, blL, blC, seq);
  finalize_loss<<<1, 32, 0, stream>>>(blL, blC, nlb, out_loss, out_cnt);
}